// SPI_68178310856797
// MI455X (gfx1250) — compile-verified
//
#include <hip/hip_runtime.h>
#include <hip/hip_bf16.h>

typedef __attribute__((ext_vector_type(2))) float v2f;
typedef __attribute__((ext_vector_type(4))) float v4f;
typedef __attribute__((ext_vector_type(8))) float v8f;
typedef __attribute__((ext_vector_type(4))) int   v4i;

#define BB   256
#define JJ   21
#define CC   256
#define HH   8
#define DD   256
#define KNN  8
#define HD   2048            // HH*DD
#define ROWS (BB*JJ)         // 5376

// GEMM tiling
#define KC      32           // K chunk staged in LDS
#define APITCH  36           // 64x32 A tile: 36r mod 64 distinct, rows 16B-aligned
#define BPPITCH 288          // (KC/2) x (128*2) pair-interleaved B tile:
                             // 288 = 32 mod 64 -> half-waves on disjoint bank sets
#define ASZ     (64*APITCH)
#define BSZ     ((KC/2)*BPPITCH)

// --- CDNA5 async global->LDS copy (ASYNCcnt path), with safe fallback ------
// Probe-verified: param0 = int4 addrspace(1)*, param1 = int4 addrspace(3)*.
#if defined(__gfx1250__) && __has_builtin(__builtin_amdgcn_global_load_async_to_lds_b128)
#define ASYNC_COPY16(gp, lp)                                                   \
    __builtin_amdgcn_global_load_async_to_lds_b128(                            \
        (__attribute__((address_space(1))) v4i*)                               \
            (__attribute__((address_space(1))) void*)(void*)(gp),              \
        (__attribute__((address_space(3))) v4i*)                               \
            (__attribute__((address_space(3))) void*)(void*)(lp),              \
        0, 0)
#if __has_builtin(__builtin_amdgcn_s_wait_asynccnt)
#define ASYNC_WAIT() __builtin_amdgcn_s_wait_asynccnt(0)
#else
#define ASYNC_WAIT() asm volatile("s_wait_asynccnt 0x0" ::: "memory")
#endif
#else
#define ASYNC_COPY16(gp, lp) (*(v4f*)(lp) = *(const v4f*)(gp))
#define ASYNC_WAIT() ((void)0)
#endif

// ---------------------------------------------------------------------------
// Pair-interleave a (K x N) matrix along K:  out[(k>>1)*2N + 2n + (k&1)] = in[k*N+n]
// Makes the WMMA B-fragment {B[k][n], B[k+1][n]} memory-contiguous.
// ---------------------------------------------------------------------------
__global__ void pack_pairs_kernel(const float* __restrict__ in,
                                  float* __restrict__ out, int K, int N)
{
    int idx = blockIdx.x * blockDim.x + threadIdx.x;
    if (idx >= K * N) return;
    int k = idx / N, n = idx - k * N;
    out[(k >> 1) * (2 * N) + n * 2 + (k & 1)] = in[idx];
}

// ---------------------------------------------------------------------------
// kNN over first 3 feature dims; stable-argsort tie-breaking, drops self.
// ---------------------------------------------------------------------------
__global__ void knn_kernel(const float* __restrict__ h, int ld,
                           int* __restrict__ nbr)
{
    int b = blockIdx.x;
    int j = threadIdx.x;
    if (j >= JJ) return;

    const float x0 = h[(b*JJ + j)*ld + 0];
    const float y0 = h[(b*JJ + j)*ld + 1];
    const float z0 = h[(b*JJ + j)*ld + 2];

    float d2[JJ];
    for (int i = 0; i < JJ; ++i) {
        float dx = h[(b*JJ + i)*ld + 0] - x0;
        float dy = h[(b*JJ + i)*ld + 1] - y0;
        float dz = h[(b*JJ + i)*ld + 2] - z0;
        d2[i] = dx*dx + dy*dy + dz*dz;
    }
    bool used[JJ];
    for (int i = 0; i < JJ; ++i) used[i] = false;

    for (int pick = 0; pick < KNN + 1; ++pick) {
        int   best = 0;
        float bd   = __builtin_inff();
        for (int i = 0; i < JJ; ++i) {
            if (!used[i] && d2[i] < bd) { bd = d2[i]; best = i; }
        }
        used[best] = true;
        if (pick >= 1) nbr[(b*JJ + j)*KNN + (pick - 1)] = best;
    }
}

// ---------------------------------------------------------------------------
// FP32 WMMA GEMM: C = A(MxK) * Bpacked(KxN)  (+= if accumulate).
// B is pair-interleaved along K (see pack_pairs_kernel); ldb = N.
// Block = 256 threads = 8 waves, block tile 64x128, wave tile 16x64,
// K chunked by 32 through double-buffered LDS fed by async global->LDS.
// Requires M%64==0, N%128==0, K%32==0 (all call sites satisfy this).
// ---------------------------------------------------------------------------
__global__ void __launch_bounds__(256)
gemm_wmma_f32(const float* __restrict__ A, int lda,
              const float* __restrict__ B, int ldb,
              float* __restrict__ C, int ldc,
              int Kdim, int accumulate)
{
    __shared__ float As[2][ASZ];
    __shared__ float Bs[2][BSZ];

    const int tid     = threadIdx.x;
    const int lane    = tid & 31;
    const int wave    = tid >> 5;
    const int waveRow = wave & 3;   // 0..3
    const int waveCol = wave >> 2;  // 0..1
    const int half    = lane >> 4;  // 0/1
    const int r       = lane & 15;

    const int rowBlk = blockIdx.x * 64;
    const int colBlk = blockIdx.y * 128;
    const int row0   = rowBlk + waveRow * 16;
    const int col0   = colBlk + waveCol * 64;

    // ---- stage one (64 x KC) A tile and (KC x 128) packed-B tile in LDS ---
    auto load_chunk = [&](int kb, int buf) {
        // A: 512 float4 chunks, 2 per thread
#pragma unroll
        for (int i = 0; i < 2; ++i) {
            const int f    = tid + i * 256;
            const int arow = f >> 3;
            const int aseg = (f & 7) << 2;
            const float* gp = A + (rowBlk + arow) * lda + kb + aseg;
            float*       lp = &As[buf][arow * APITCH + aseg];
            ASYNC_COPY16(gp, lp);
        }
        // B: 16 kpair-rows x 256 floats = 1024 float4 chunks, 4 per thread
#pragma unroll
        for (int i = 0; i < 4; ++i) {
            const int f     = tid + i * 256;
            const int bkrow = f >> 6;          // 0..15
            const int bseg  = (f & 63) << 2;   // float offset in row
            const float* gp = B + ((kb >> 1) + bkrow) * (2 * ldb)
                                + colBlk * 2 + bseg;
            float*       lp = &Bs[buf][bkrow * BPPITCH + bseg];
            ASYNC_COPY16(gp, lp);
        }
    };

    v8f acc[4];
    if (accumulate) {
        for (int t = 0; t < 4; ++t)
            for (int i = 0; i < 8; ++i)
                acc[t][i] = C[(row0 + i + 8*half) * ldc + col0 + t*16 + r];
    } else {
        for (int t = 0; t < 4; ++t)
            for (int i = 0; i < 8; ++i)
                acc[t][i] = 0.0f;
    }

    const int nchunks = Kdim / KC;
    load_chunk(0, 0);
    int cur = 0;

    const int rowA = (waveRow * 16 + r) * APITCH + 2 * half;       // A frag base
    const int colb = half * BPPITCH + (waveCol * 64 + r) * 2;      // B frag base

    for (int c = 0; c < nchunks; ++c) {
        ASYNC_WAIT();           // own async ops for chunk c complete
        __syncthreads();        // all waves' chunk-c data visible
        if (c + 1 < nchunks) load_chunk((c + 1) * KC, cur ^ 1);

        const float* __restrict__ Arow  = &As[cur][rowA];
        const float* __restrict__ Bbase = &Bs[cur][colb];
#pragma unroll
        for (int kloc = 0; kloc < KC; kloc += 4) {
            v2f a = *(const v2f*)(Arow + kloc);                 // ds_load_b64
            const float* bp = Bbase + (kloc >> 1) * BPPITCH;
#pragma unroll
            for (int t = 0; t < 4; ++t) {
                v2f bfrag = *(const v2f*)(bp + t * 32);         // ds_load_b64
                acc[t] = __builtin_amdgcn_wmma_f32_16x16x4_f32(
                    false, a, false, bfrag, (short)0, acc[t], false, false);
            }
        }
        __syncthreads();        // done reading buf[cur] before chunk c+2 reuses it
        cur ^= 1;
    }

    for (int t = 0; t < 4; ++t)
        for (int i = 0; i < 8; ++i)
            C[(row0 + i + 8*half) * ldc + col0 + t*16 + r] = acc[t][i];
}

// ---------------------------------------------------------------------------
// Attention: one wave per (b, j, head); wave32 shuffle reductions.
// Output overwrites Q in place (each wave reads only its own Q slice).
// ---------------------------------------------------------------------------
__global__ void __launch_bounds__(256)
attn_kernel(float* __restrict__ q,
            const float* __restrict__ k,
            const float* __restrict__ v,
            const int* __restrict__ nbr)
{
    const int lane = threadIdx.x & 31;
    const int wave = threadIdx.x >> 5;
    const int unit = blockIdx.x * 8 + wave;          // < BB*JJ*HH
    if (unit >= BB*JJ*HH) return;

    const int hh = unit % HH;
    const int bj = unit / HH;
    const int b  = bj / JJ;

    float* qp = q + bj * HD + hh * DD;
    float qv[8];
#pragma unroll
    for (int t = 0; t < 8; ++t) qv[t] = qp[lane + 32*t];

    const int* np = nbr + bj * KNN;

    float logits[KNN];
    for (int kk = 0; kk < KNN; ++kk) {
        const int n = np[kk];
        const float* kp = k + (b*JJ + n) * HD + hh * DD;
        float s = 0.0f;
#pragma unroll
        for (int t = 0; t < 8; ++t) s += qv[t] * kp[lane + 32*t];
        for (int m = 16; m >= 1; m >>= 1) s += __shfl_xor(s, m, 32);
        s *= (1.0f / 16.0f);                          // 1/sqrt(D)
        logits[kk] = (s > 0.0f) ? s : 0.2f * s;       // leaky_relu
    }

    float mx = logits[0];
    for (int kk = 1; kk < KNN; ++kk) mx = fmaxf(mx, logits[kk]);
    float w[KNN], denom = 0.0f;
    for (int kk = 0; kk < KNN; ++kk) { w[kk] = __expf(logits[kk] - mx); denom += w[kk]; }
    const float inv = 1.0f / denom;

    float out[8];
#pragma unroll
    for (int t = 0; t < 8; ++t) out[t] = 0.0f;
    for (int kk = 0; kk < KNN; ++kk) {
        const int n = np[kk];
        const float* vp = v + (b*JJ + n) * HD + hh * DD;
        const float a = w[kk] * inv;
#pragma unroll
        for (int t = 0; t < 8; ++t) out[t] += a * vp[lane + 32*t];
    }
#pragma unroll
    for (int t = 0; t < 8; ++t) qp[lane + 32*t] = out[t];
}

// ---------------------------------------------------------------------------
__global__ void relu_stats_kernel(float* __restrict__ h, float* __restrict__ stat, int n)
{
    int i = blockIdx.x * blockDim.x + threadIdx.x;
    float x = 0.0f;
    if (i < n) {
        x = h[i];
        x = (x > 0.0f) ? x : 0.0f;
        h[i] = x;
    }
    for (int m = 16; m >= 1; m >>= 1) x += __shfl_xor(x, m, 32);
    if ((threadIdx.x & 31) == 0) atomicAdd(stat, x);
}

__global__ void init_stats_kernel(float* __restrict__ stat)
{
    if (threadIdx.x < 2) stat[threadIdx.x] = 0.0f;
}

__global__ void final_kernel(const float* __restrict__ h,
                             const float* __restrict__ fc_w,
                             const float* __restrict__ fc_b,
                             const float* __restrict__ stat,
                             float* __restrict__ out)
{
    const int r = blockIdx.x * blockDim.x + threadIdx.x;
    if (r < ROWS) {
        float c0 = fc_b[0], c1 = fc_b[1], c2 = fc_b[2];
        const float* hp = h + r * DD;
        for (int d = 0; d < DD; ++d) {
            const float x = hp[d];
            c0 += x * fc_w[d*3 + 0];
            c1 += x * fc_w[d*3 + 1];
            c2 += x * fc_w[d*3 + 2];
        }
        out[r*3 + 0] = c0;
        out[r*3 + 1] = c1;
        out[r*3 + 2] = c2;
    }
    if (r == 0) {
        const float cnt = (float)ROWS * (float)DD;
        out[ROWS*3] = 0.5f * (stat[0] + stat[1]) / cnt;
    }
}

// ---------------------------------------------------------------------------
extern "C" void kernel_launch(void* const* d_in, const int* in_sizes, int n_in,
                              void* d_out, int out_size, void* d_ws, size_t ws_size,
                              hipStream_t stream)
{
    const float* joint_feats = (const float*)d_in[0];   // (B,J,C)
    const float* wq     = (const float*)d_in[1];        // (L,C,HD)
    const float* wk     = (const float*)d_in[2];        // (L,C,HD)
    const float* wv     = (const float*)d_in[3];        // (L,C,HD)
    const float* wp     = (const float*)d_in[4];        // (L,HD,D)
    const float* wsw    = (const float*)d_in[5];        // (L,C,D)
    const float* wskip0 = (const float*)d_in[6];        // (C,D)
    const float* fc_w   = (const float*)d_in[7];        // (D,3)
    const float* fc_b   = (const float*)d_in[8];        // (3,)
    float* outp = (float*)d_out;

    // --- workspace carve-out (256B aligned) ---
    char* p = (char*)d_ws;
    auto alloc = [&](size_t bytes) -> void* {
        void* r = (void*)p;
        p += (bytes + 255) & ~(size_t)255;
        return r;
    };
    float* stats = (float*)alloc(2 * sizeof(float));
    int*   nbr   = (int*)  alloc((size_t)ROWS * KNN * sizeof(int));
    float* h1    = (float*)alloc((size_t)ROWS * DD * sizeof(float));
    float* h2    = (float*)alloc((size_t)ROWS * DD * sizeof(float));
    float* qb    = (float*)alloc((size_t)ROWS * HD * sizeof(float));
    float* kb    = (float*)alloc((size_t)ROWS * HD * sizeof(float));
    float* vb    = (float*)alloc((size_t)ROWS * HD * sizeof(float));
    // pair-interleaved weights
    float* wq_p  = (float*)alloc((size_t)2*CC*HD * sizeof(float));
    float* wk_p  = (float*)alloc((size_t)2*CC*HD * sizeof(float));
    float* wv_p  = (float*)alloc((size_t)2*CC*HD * sizeof(float));
    float* wp_p  = (float*)alloc((size_t)2*HD*DD * sizeof(float));
    float* wsw_p = (float*)alloc((size_t)2*CC*DD * sizeof(float));
    float* wsk_p = (float*)alloc((size_t)CC*DD   * sizeof(float));
    (void)ws_size; (void)n_in; (void)in_sizes; (void)out_size;

    auto pack = [&](const float* in, float* out, int K, int N) {
        int total = K * N;
        hipLaunchKernelGGL(pack_pairs_kernel, dim3((total + 255) / 256), dim3(256),
                           0, stream, in, out, K, N);
    };
    auto gemm = [&](const float* A, int lda, const float* B, int ldb,
                    float* C, int ldc, int M, int N, int Kd, int accum) {
        dim3 grid(M / 64, N / 128);
        hipLaunchKernelGGL(gemm_wmma_f32, grid, dim3(256), 0, stream,
                           A, lda, B, ldb, C, ldc, Kd, accum);
    };

    hipLaunchKernelGGL(init_stats_kernel, dim3(1), dim3(32), 0, stream, stats);

    // one-time (per launch) weight repack; K dims even so layer bases persist
    pack(wq,     wq_p,  2*CC, HD);
    pack(wk,     wk_p,  2*CC, HD);
    pack(wv,     wv_p,  2*CC, HD);
    pack(wp,     wp_p,  2*HD, DD);
    pack(wsw,    wsw_p, 2*CC, DD);
    pack(wskip0, wsk_p, CC,   DD);

    const float* h  = joint_feats;   // layer input
    float*       hn = h1;            // layer output
    for (int l = 0; l < 2; ++l) {
        // dynamic kNN graph on first 3 dims of current features
        hipLaunchKernelGGL(knn_kernel, dim3(BB), dim3(32), 0, stream, h, CC, nbr);

        // Q/K/V projections: (5376x256) @ (256x2048)
        gemm(h, CC, wq_p + (size_t)l*CC*HD, HD, qb, HD, ROWS, HD, CC, 0);
        gemm(h, CC, wk_p + (size_t)l*CC*HD, HD, kb, HD, ROWS, HD, CC, 0);
        gemm(h, CC, wv_p + (size_t)l*CC*HD, HD, vb, HD, ROWS, HD, CC, 0);

        // neighborhood attention (output overwrites qb)
        hipLaunchKernelGGL(attn_kernel, dim3(ROWS), dim3(256), 0, stream,
                           qb, kb, vb, nbr);

        // hn = attn_out @ wp + h @ ws[l] + h0 @ wskip0
        gemm(qb, HD, wp_p  + (size_t)l*HD*DD, DD, hn, DD, ROWS, DD, HD, 0);
        gemm(h,  CC, wsw_p + (size_t)l*CC*DD, DD, hn, DD, ROWS, DD, CC, 1);
        gemm(joint_feats, CC, wsk_p, DD, hn, DD, ROWS, DD, CC, 1);

        // hn = relu(hn); stats[l] += sum(hn)
        hipLaunchKernelGGL(relu_stats_kernel, dim3(ROWS), dim3(256), 0, stream,
                           hn, stats + l, ROWS * DD);

        h  = hn;
        hn = (l == 0) ? h2 : h1;
    }

    // coords + alphas
    hipLaunchKernelGGL(final_kernel, dim3((ROWS + 255) / 256), dim3(256), 0, stream,
                       h, fc_w, fc_b, stats, outp);
}